// GTM_SM_44513041056116
// MI455X (gfx1250) — compile-verified
//
#include <hip/hip_runtime.h>
#include <math.h>

#define B_    1024
#define T_    288
#define OBS_  256
#define A_    5
#define S_    2
#define Z_    16
#define PRED_ (T_ - OBS_)

typedef __attribute__((ext_vector_type(2))) float v2f;
typedef __attribute__((ext_vector_type(8))) float v8f;
typedef __attribute__((ext_vector_type(4))) unsigned int v4u;
typedef __attribute__((ext_vector_type(8))) int v8i;
typedef __attribute__((ext_vector_type(4))) int v4i;

// ---- d_out layout (concatenated flat, reference return order) ----
#define OFF_NLL     0
#define OFF_ST_OBS  1
#define OFF_ST_PRED (OFF_ST_OBS + OBS_*B_*S_)
#define OFF_ZM_OBS  (OFF_ST_PRED + PRED_*B_*S_)
#define OFF_ZS_OBS  (OFF_ZM_OBS + OBS_*B_*Z_)
#define OFF_XREC    (OFF_ZS_OBS + OBS_*B_*Z_)

// =====================================================================
// TDM helper: 1-D tensor_load_to_lds of `nwords` f32 from `gaddr` to the
// LDS offset `lds_off`. D# built per CDNA5 ISA:
//   group0: count=1 | lds_addr | global_addr[56:0] | type=2
//   group1: data_size=4B, tensor_dim0=tile_dim0=nwords, dim1=1,
//           dim0_stride=nwords, everything else 0 (no pad/iterate/mcast)
// Issue from one wave only (TDM ignores EXEC; issues once per wave).
// =====================================================================
__device__ static inline void tdm_load_f32_1d(unsigned int lds_off,
                                              const void* gaddr,
                                              unsigned int nwords) {
  unsigned long long ga = (unsigned long long)gaddr;
  v4u g0;
  g0.x = 1u;                                   // count=1, user mode
  g0.y = lds_off;                              // lds_addr (bytes)
  g0.z = (unsigned int)(ga & 0xffffffffu);     // global_addr[31:0]
  g0.w = (unsigned int)((ga >> 32) & 0x01ffffffu) | (2u << 30); // [56:32]|type=2
  v8i g1 = {0, 0, 0, 0, 0, 0, 0, 0};
  g1[0] = (int)(2u << 16);                     // data_size = 2 (4 bytes)
  g1[1] = (int)((nwords & 0xffffu) << 16);     // tensor_dim0[15:0]  (bits 63:48)
  g1[2] = (int)(((nwords >> 16) & 0xffffu)     // tensor_dim0[31:16] (bits 79:64)
                | (1u << 16));                 // tensor_dim1 = 1    (bits 95:80)
  g1[3] = (int)((nwords & 0xffffu) << 16);     // tile_dim0 (bits 127:112)
  g1[5] = (int)nwords;                         // tensor_dim0_stride[31:0]
  v4i z4 = {0, 0, 0, 0};
#if __has_include(<hip/amd_detail/amd_gfx1250_TDM.h>)
  v8i z8 = {0, 0, 0, 0, 0, 0, 0, 0};
  __builtin_amdgcn_tensor_load_to_lds(g0, g1, z4, z4, z8, 0);   // clang-23 form
#else
  __builtin_amdgcn_tensor_load_to_lds(g0, g1, z4, z4, 0);       // ROCm 7.2 form
#endif
}

// =====================================================================
// Kernel 1: sequential state-space scan. One batch element per thread,
// 287 sequential steps. Also zeroes the NLL accumulator.
// =====================================================================
__global__ __launch_bounds__(128) void k_scan(
    const float* __restrict__ aoh,   // (B, A, T)
    const float* __restrict__ st0,   // (B, S)
    const float* __restrict__ noise, // (T, B, S)
    const float* __restrict__ Wst,   // (S, A)
    const float* __restrict__ w1,    // (5, S)
    const float* __restrict__ b1,    // (5,)
    const float* __restrict__ w2,    // (S, 5)
    const float* __restrict__ b2,    // (S,)
    float* __restrict__ out) {
  int b = blockIdx.x * blockDim.x + threadIdx.x;
  if (b == 0) out[OFF_NLL] = 0.0f;
  if (b >= B_) return;

  float W[2][5], sw1[5][2], sb1[5], sw2[2][5], sb2[2];
  #pragma unroll
  for (int s = 0; s < 2; ++s)
    #pragma unroll
    for (int a = 0; a < 5; ++a) W[s][a] = Wst[s*5 + a];
  #pragma unroll
  for (int j = 0; j < 5; ++j) {
    sw1[j][0] = w1[j*2 + 0]; sw1[j][1] = w1[j*2 + 1]; sb1[j] = b1[j];
  }
  #pragma unroll
  for (int s = 0; s < 2; ++s) {
    #pragma unroll
    for (int j = 0; j < 5; ++j) sw2[s][j] = w2[s*5 + j];
    sb2[s] = b2[s];
  }

  float s0 = st0[b*2 + 0], s1 = st0[b*2 + 1];
  out[OFF_ST_OBS + (size_t)b*2 + 0] = s0;   // st_obs[0] = st0
  out[OFF_ST_OBS + (size_t)b*2 + 1] = s1;

  for (int t = 1; t < T_; ++t) {
    float m0 = 0.0f, m1 = 0.0f;
    #pragma unroll
    for (int a = 0; a < 5; ++a) {
      float v = aoh[((size_t)b*A_ + a)*T_ + t];
      m0 += v * W[0][a];
      m1 += v * W[1][a];
    }
    float p0 = s0 + m0, p1 = s1 + m1;
    float q0 = sb2[0], q1 = sb2[1];
    #pragma unroll
    for (int j = 0; j < 5; ++j) {
      float hj = tanhf(p0*sw1[j][0] + p1*sw1[j][1] + sb1[j]);
      q0 += hj * sw2[0][j];
      q1 += hj * sw2[1][j];
    }
    float g0 = 1.0f / (1.0f + expf(-q0));
    float g1 = 1.0f / (1.0f + expf(-q1));
    s0 = s0 + m0*g0 + noise[((size_t)t*B_ + b)*2 + 0];
    s1 = s1 + m1*g1 + noise[((size_t)t*B_ + b)*2 + 1];
    if (t < OBS_) {
      out[OFF_ST_OBS + ((size_t)t*B_ + b)*2 + 0] = s0;
      out[OFF_ST_OBS + ((size_t)t*B_ + b)*2 + 1] = s1;
    } else {
      int k = t - OBS_;
      out[OFF_ST_PRED + ((size_t)k*B_ + b)*2 + 0] = s0;
      out[OFF_ST_PRED + ((size_t)k*B_ + b)*2 + 1] = s1;
    }
  }
}

// =====================================================================
// Kernel 2: patch gather + conv3x3 + relu + maxpool2x2 -> h[72] in LDS,
// then fp32 WMMA GEMM (128x72)*(72x32) for [mean|std] heads.
// mean_w/std_w staged to LDS by the Tensor Data Mover.
// 128 threads (4 waves) per block, 128 samples per block.
// =====================================================================
__global__ __launch_bounds__(128) void k_encode(
    const float* __restrict__ x,      // (B,3,32,32)
    const int*   __restrict__ pos,    // (B,2,T)
    const float* __restrict__ conv_w, // (8,3,3,3)
    const float* __restrict__ conv_b, // (8,)
    const float* __restrict__ mean_w, // (Z,72)
    const float* __restrict__ mean_b, // (Z,)
    const float* __restrict__ std_w,  // (Z,72)
    const float* __restrict__ std_b,  // (Z,)
    float* __restrict__ out,
    float* __restrict__ zm_pred,      // ws: (PRED*B, Z)
    float* __restrict__ zs_pred) {    // ws: (PRED*B, Z)
  __shared__ float sh_h[128][73];     // activations h (padded)
  __shared__ float sh_w[72][32];      // [mean_w^T | std_w^T]
  __shared__ float sh_cw[216];        // conv weights
  __shared__ float sh_stage[2][16*72]; // raw TDM staging: mean_w, std_w

  int tid = threadIdx.x;
  int s = blockIdx.x * 128 + tid;       // flat sample t*B + b
  int t = s / B_, b = s - t * B_;
  int ph = pos[((size_t)b*2 + 0)*T_ + t];
  int pw = pos[((size_t)b*2 + 1)*T_ + t];
  const float* xb = x + (size_t)b * 3 * 32 * 32;

  // Prefetch the patch cachelines while weights are being staged.
  #pragma unroll
  for (int c = 0; c < 3; ++c)
    __builtin_prefetch(xb + ((size_t)c*32 + 3*ph)*32 + 3*pw, 0, 3);

  // Async TDM load of the two head matrices into LDS (wave 0 issues once).
  if (tid < 32) {
    unsigned int base = (unsigned int)(unsigned long long)(&sh_stage[0][0]);
    tdm_load_f32_1d(base,            mean_w, 16*72);
    tdm_load_f32_1d(base + 16*72*4,  std_w,  16*72);
    __builtin_amdgcn_s_wait_tensorcnt(0);
  }
  for (int i = tid; i < 216; i += 128) sh_cw[i] = conv_w[i];
  __syncthreads();

  // Repack staged weights into the B-matrix layout [K=72][N=32].
  for (int i = tid; i < 72*32; i += 128) {
    int k = i >> 5, n = i & 31;
    sh_w[k][n] = (n < 16) ? sh_stage[0][n*72 + k] : sh_stage[1][(n - 16)*72 + k];
  }

  // Gather 3x8x8 patch into registers (constant-indexed via full unroll).
  float p[3][8][8];
  #pragma unroll
  for (int c = 0; c < 3; ++c)
    #pragma unroll
    for (int r = 0; r < 8; ++r) {
      const float* src = xb + ((size_t)c*32 + 3*ph + r)*32 + 3*pw;
      #pragma unroll
      for (int q = 0; q < 8; ++q) p[c][r][q] = src[q];
    }
  __syncthreads();

  // conv + relu + pool, per output channel (ch loop NOT unrolled).
  for (int ch = 0; ch < 8; ++ch) {
    float wr[27];
    #pragma unroll
    for (int i = 0; i < 27; ++i) wr[i] = sh_cw[ch*27 + i];
    float cbv = conv_b[ch];
    #pragma unroll
    for (int py = 0; py < 3; ++py)
      #pragma unroll
      for (int px = 0; px < 3; ++px) {
        float mx = 0.0f;                 // relu outputs are >= 0
        #pragma unroll
        for (int dy = 0; dy < 2; ++dy)
          #pragma unroll
          for (int dx = 0; dx < 2; ++dx) {
            float acc = cbv;
            #pragma unroll
            for (int c = 0; c < 3; ++c)
              #pragma unroll
              for (int ky = 0; ky < 3; ++ky)
                #pragma unroll
                for (int kx = 0; kx < 3; ++kx)
                  acc += p[c][2*py + dy + ky][2*px + dx + kx] *
                         wr[(c*3 + ky)*3 + kx];
            mx = fmaxf(mx, acc);
          }
        sh_h[tid][ch*9 + py*3 + px] = mx;
      }
  }
  __syncthreads();

  // --- WMMA: (16x72) x (72x32) per tile, 8 tiles, 2 per wave ---
  int wave = tid >> 5, lane = tid & 31;
  int lhalf = lane >> 4;      // K sub-half for A/B fragments
  int l16   = lane & 15;
  #pragma unroll
  for (int rt = 0; rt < 2; ++rt) {
    int tile = wave * 2 + rt;
    int row  = tile * 16 + l16;          // A-matrix row (sample in block)
    v8f acc0 = {0,0,0,0,0,0,0,0};        // mean head (N=0..15)
    v8f acc1 = {0,0,0,0,0,0,0,0};        // std head  (N=16..31)
    for (int kk = 0; kk < 18; ++kk) {    // K = 72 = 18 * 4
      int kb = kk*4 + lhalf*2;
      v2f a;  a.x  = sh_h[row][kb];      a.y  = sh_h[row][kb + 1];
      v2f b0; b0.x = sh_w[kb][l16];      b0.y = sh_w[kb + 1][l16];
      v2f b1; b1.x = sh_w[kb][16 + l16]; b1.y = sh_w[kb + 1][16 + l16];
      acc0 = __builtin_amdgcn_wmma_f32_16x16x4_f32(
          false, a, false, b0, (short)0, acc0, false, false);
      acc1 = __builtin_amdgcn_wmma_f32_16x16x4_f32(
          false, a, false, b1, (short)0, acc1, false, false);
    }
    int n = l16;
    float mb = mean_b[n];
    float sb = std_b[n];
    int sbase = blockIdx.x * 128 + tile * 16;
    #pragma unroll
    for (int v = 0; v < 8; ++v) {
      int M = v + lhalf * 8;             // C/D layout row
      int sidx = sbase + M;
      float mval = acc0[v] + mb;
      float sval = expf(acc1[v] + sb);
      if (sidx < OBS_*B_) {
        out[OFF_ZM_OBS + (size_t)sidx*Z_ + n] = mval;
        out[OFF_ZS_OBS + (size_t)sidx*Z_ + n] = sval;
      } else {
        int sp = sidx - OBS_*B_;
        zm_pred[(size_t)sp*Z_ + n] = mval;
        zs_pred[(size_t)sp*Z_ + n] = sval;
      }
    }
  }
}

// =====================================================================
// Kernel 3: z = mu + sigma*eps; WMMA GEMM (128x16)*(16x80 padded) ->
// tanh decoder; per-thread 4x4 stride-2 conv-transpose; NLL reduction.
// dec_w staged to LDS by the Tensor Data Mover.
// =====================================================================
__global__ __launch_bounds__(128) void k_decode(
    const float* __restrict__ x,        // (B,3,32,32)
    const int*   __restrict__ pos,      // (B,2,T)
    const float* __restrict__ eps,      // (PRED,B,Z)
    const float* __restrict__ dec_w,    // (72,Z)
    const float* __restrict__ dec_b,    // (72,)
    const float* __restrict__ dcw,      // (8,3,4,4)
    const float* __restrict__ dcb_g,    // (3,)
    const float* __restrict__ zm_pred,
    const float* __restrict__ zs_pred,
    float* __restrict__ out) {
  __shared__ float sh_z[128][16];
  __shared__ float sh_dw[16][80];   // dec_w^T zero-padded to N=80
  __shared__ float sh_d[128][80];
  __shared__ float sh_dcw[384];
  __shared__ float sh_red[128];
  __shared__ float sh_stage[72*16]; // raw TDM staging of dec_w

  int tid = threadIdx.x;
  int s = blockIdx.x * 128 + tid;   // pred-flat sample (t-OBS)*B + b

  if (tid < 32) {
    unsigned int base = (unsigned int)(unsigned long long)(&sh_stage[0]);
    tdm_load_f32_1d(base, dec_w, 72*16);
    __builtin_amdgcn_s_wait_tensorcnt(0);
  }
  for (int i = tid; i < 384; i += 128) sh_dcw[i] = dcw[i];
  #pragma unroll
  for (int j = 0; j < 16; ++j)
    sh_z[tid][j] = zm_pred[(size_t)s*Z_ + j] +
                   zs_pred[(size_t)s*Z_ + j] * eps[(size_t)s*Z_ + j];
  __syncthreads();
  for (int i = tid; i < 16*80; i += 128) {
    int k = i / 80, n = i - k*80;
    sh_dw[k][n] = (n < 72) ? sh_stage[n*Z_ + k] : 0.0f;
  }
  __syncthreads();

  // --- WMMA decoder GEMM ---
  int wave = tid >> 5, lane = tid & 31, lhalf = lane >> 4, l16 = lane & 15;
  #pragma unroll
  for (int rt = 0; rt < 2; ++rt) {
    int tile = wave * 2 + rt;
    int row  = tile * 16 + l16;
    for (int nt = 0; nt < 5; ++nt) {
      v8f acc = {0,0,0,0,0,0,0,0};
      #pragma unroll
      for (int kk = 0; kk < 4; ++kk) {   // K = 16 = 4 * 4
        int kb = kk*4 + lhalf*2;
        v2f a;  a.x  = sh_z[row][kb];          a.y  = sh_z[row][kb + 1];
        v2f bb; bb.x = sh_dw[kb][nt*16 + l16]; bb.y = sh_dw[kb + 1][nt*16 + l16];
        acc = __builtin_amdgcn_wmma_f32_16x16x4_f32(
            false, a, false, bb, (short)0, acc, false, false);
      }
      int n = nt*16 + l16;
      float db = (n < 72) ? dec_b[n] : 0.0f;
      #pragma unroll
      for (int v = 0; v < 8; ++v) {
        int M = v + lhalf * 8;
        if (n < 72) sh_d[tile*16 + M][n] = tanhf(acc[v] + db);
      }
    }
  }
  __syncthreads();

  // --- conv-transpose (transpose_kernel => spatially flipped) + loss ---
  int t = OBS_ + s / B_, b = s % B_;
  int ph = pos[((size_t)b*2 + 0)*T_ + t];
  int pw = pos[((size_t)b*2 + 1)*T_ + t];
  const float* xb = x + (size_t)b * 3 * 32 * 32;
  const float* dd = sh_d[tid];
  float loss = 0.0f;
  float cb0 = dcb_g[0], cb1 = dcb_g[1], cb2 = dcb_g[2];

  for (int o = 0; o < 3; ++o) {
    float bias = (o == 0) ? cb0 : (o == 1) ? cb1 : cb2;
    for (int oy = 0; oy < 8; ++oy)
      for (int ox = 0; ox < 8; ++ox) {
        float acc = bias;
        #pragma unroll
        for (int ky = 0; ky < 4; ++ky) {
          int yy = oy - ky;
          if (yy < 0 || (yy & 1) || yy > 4) continue;
          int y = yy >> 1;
          #pragma unroll
          for (int kx = 0; kx < 4; ++kx) {
            int xx = ox - kx;
            if (xx < 0 || (xx & 1) || xx > 4) continue;
            int xq = xx >> 1;
            #pragma unroll
            for (int i = 0; i < 8; ++i)
              acc += dd[i*9 + y*3 + xq] *
                     sh_dcw[((i*3 + o)*4 + (3 - ky))*4 + (3 - kx)];
          }
        }
        float pv = xb[((size_t)o*32 + 3*ph + oy)*32 + 3*pw + ox];
        float df = acc - pv;
        loss += df * df;
        out[OFF_XREC + (size_t)s*192 + o*64 + oy*8 + ox] = acc;
      }
  }
  sh_red[tid] = loss;
  __syncthreads();
  #pragma unroll
  for (int off = 64; off > 0; off >>= 1) {
    if (tid < off) sh_red[tid] += sh_red[tid + off];
    __syncthreads();
  }
  if (tid == 0) atomicAdd(&out[OFF_NLL], sh_red[0]);
}

// =====================================================================
extern "C" void kernel_launch(void* const* d_in, const int* in_sizes, int n_in,
                              void* d_out, int out_size, void* d_ws, size_t ws_size,
                              hipStream_t stream) {
  const float* x     = (const float*)d_in[0];
  const float* aoh   = (const float*)d_in[1];
  const int*   pos   = (const int*)  d_in[2];
  const float* st0   = (const float*)d_in[3];
  const float* noise = (const float*)d_in[4];
  const float* eps   = (const float*)d_in[5];
  const float* Wst   = (const float*)d_in[6];
  const float* sw1   = (const float*)d_in[7];
  const float* sb1   = (const float*)d_in[8];
  const float* sw2   = (const float*)d_in[9];
  const float* sb2   = (const float*)d_in[10];
  const float* cw    = (const float*)d_in[11];
  const float* cb    = (const float*)d_in[12];
  const float* mw    = (const float*)d_in[13];
  const float* mb    = (const float*)d_in[14];
  const float* stw   = (const float*)d_in[15];
  const float* stb   = (const float*)d_in[16];
  const float* dw    = (const float*)d_in[17];
  const float* db    = (const float*)d_in[18];
  const float* dcw   = (const float*)d_in[19];
  const float* dcb   = (const float*)d_in[20];
  float* out = (float*)d_out;
  float* zm_pred = (float*)d_ws;                 // PRED*B*Z floats
  float* zs_pred = zm_pred + (size_t)PRED_*B_*Z_;

  k_scan<<<(B_ + 127)/128, 128, 0, stream>>>(aoh, st0, noise, Wst,
                                             sw1, sb1, sw2, sb2, out);
  k_encode<<<(T_*B_)/128, 128, 0, stream>>>(x, pos, cw, cb, mw, mb, stw, stb,
                                            out, zm_pred, zs_pred);
  k_decode<<<(PRED_*B_)/128, 128, 0, stream>>>(x, pos, eps, dw, db, dcw, dcb,
                                               zm_pred, zs_pred, out);
}